// GCLMEncoder_53326313947266
// MI455X (gfx1250) — compile-verified
//
#include <hip/hip_runtime.h>
#include <hip/hip_bf16.h>

typedef __attribute__((ext_vector_type(2))) float v2f;
typedef __attribute__((ext_vector_type(8))) float v8f;

#define HID 128

// ---------------- degree / normalization ----------------

__global__ void k_deg_init(float* __restrict__ deg, int n) {
    int i = blockIdx.x * blockDim.x + threadIdx.x;
    if (i < n) deg[i] = 1.0f;  // self loop
}

__global__ void k_deg_count(const int* __restrict__ dst, float* __restrict__ deg, int E) {
    int e = blockIdx.x * blockDim.x + threadIdx.x;
    if (e < E) atomicAdd(&deg[dst[e]], 1.0f);
}

__global__ void k_dinv(float* __restrict__ deg, int n) {
    int i = blockIdx.x * blockDim.x + threadIdx.x;
    if (i < n) deg[i] = rsqrtf(deg[i]);  // deg >= 1 always
}

// ---------------- h = x @ W^T via V_WMMA_F32_16X16X4_F32 ----------------
// Block: 256 threads = 8 waves. Each wave: 16 rows x all 128 cols.
// W (128x128 f32, 64KB) staged in LDS once per block.

__global__ void k_gemm_wmma(const float* __restrict__ x, const float* __restrict__ W,
                            float* __restrict__ h, int nrows) {
    __shared__ float Wlds[HID * HID];
    int tid = threadIdx.x;
    #pragma unroll 4
    for (int i = tid; i < HID * HID; i += 256) Wlds[i] = W[i];
    __syncthreads();

    int wave = tid >> 5;
    int lane = tid & 31;
    int row0 = (blockIdx.x * 8 + wave) * 16;
    if (row0 >= nrows) return;  // wave-uniform: EXEC stays all ones for WMMA

    int lhalf = lane & 15;
    int khi   = (lane >> 4) << 1;  // 0 for lanes 0-15, 2 for lanes 16-31

    v8f acc[8];
    #pragma unroll
    for (int t = 0; t < 8; ++t) acc[t] = (v8f){};

    for (int k = 0; k < HID; k += 4) {
        // A fragment (16x4): lane L<16 -> x[row0+L][k+0 / k+1]; L>=16 -> k+2 / k+3
        v2f a = *(const v2f*)(x + (size_t)(row0 + lhalf) * HID + k + khi);
        #pragma unroll
        for (int t = 0; t < 8; ++t) {
            // B fragment (4x16): B[k'][j] = W[j][k']; lanes index N (column j)
            v2f bf = *(const v2f*)(Wlds + (t * 16 + lhalf) * HID + k + khi);
            acc[t] = __builtin_amdgcn_wmma_f32_16x16x4_f32(
                false, a, false, bf, (short)0, acc[t], false, false);
        }
    }

    // D layout: VGPR v -> M = v (lanes<16) / v+8 (lanes>=16); N = lane&15
    int mhi = (lane < 16) ? 0 : 8;
    int col = lane & 15;
    #pragma unroll
    for (int v = 0; v < 8; ++v) {
        float* hp = h + (size_t)(row0 + v + mhi) * HID + col;
        #pragma unroll
        for (int t = 0; t < 8; ++t) hp[t * 16] = acc[t][v];
    }
}

// ---------------- self-loop term: out = h * dinv^2 ----------------

__global__ void k_self(const float* __restrict__ h, const float* __restrict__ dinv,
                       float* __restrict__ out, int total) {
    int idx = blockIdx.x * blockDim.x + threadIdx.x;
    if (idx < total) {
        int node = idx >> 7;  // /128
        float d = dinv[node];
        out[idx] = h[idx] * d * d;
    }
}

// ---------------- edge scatter: one wave per edge ----------------

__global__ void k_scatter(const int* __restrict__ src, const int* __restrict__ dst,
                          const float* __restrict__ h, const float* __restrict__ dinv,
                          float* __restrict__ out, int E) {
    int gid  = blockIdx.x * blockDim.x + threadIdx.x;
    int e    = gid >> 5;
    int lane = gid & 31;
    if (e >= E) return;
    int s = src[e];
    int d = dst[e];
    float norm = dinv[s] * dinv[d];
    const float4* hp = (const float4*)(h + (size_t)s * HID);
    float4 v = hp[lane];                       // 32 lanes x 16B = full 128-float row
    float* op = out + (size_t)d * HID + lane * 4;
    atomicAdd(op + 0, v.x * norm);
    atomicAdd(op + 1, v.y * norm);
    atomicAdd(op + 2, v.z * norm);
    atomicAdd(op + 3, v.w * norm);
}

// ---------------- bias + PReLU ----------------

__global__ void k_final(float* __restrict__ out, const float* __restrict__ b,
                        const float* __restrict__ prelu_a, int total) {
    int idx = blockIdx.x * blockDim.x + threadIdx.x;
    if (idx < total) {
        float v = out[idx] + b[idx & (HID - 1)];
        out[idx] = (v >= 0.0f) ? v : prelu_a[0] * v;
    }
}

extern "C" void kernel_launch(void* const* d_in, const int* in_sizes, int n_in,
                              void* d_out, int out_size, void* d_ws, size_t ws_size,
                              hipStream_t stream) {
    const float* x       = (const float*)d_in[0];
    const int*   ei      = (const int*)d_in[1];   // [2, E] int
    const float* W       = (const float*)d_in[2];
    const float* bias    = (const float*)d_in[3];
    const float* prelu_a = (const float*)d_in[4];
    float* out = (float*)d_out;

    const int n = in_sizes[0] / HID;     // 100000
    const int E = in_sizes[1] / 2;       // 1600000
    const int* src = ei;
    const int* dst = ei + E;

    // workspace: h [n*128 floats] then deg/dinv [n floats]
    float* h   = (float*)d_ws;
    float* deg = h + (size_t)n * HID;

    const int T = 256;
    // 1) degree with self loops -> dinv (in place)
    k_deg_init <<<(n + T - 1) / T, T, 0, stream>>>(deg, n);
    k_deg_count<<<(E + T - 1) / T, T, 0, stream>>>(dst, deg, E);
    k_dinv     <<<(n + T - 1) / T, T, 0, stream>>>(deg, n);

    // 2) h = x @ W^T  (WMMA f32 16x16x4); 16 rows per wave, 8 waves per block
    int row_tiles = (n + 15) / 16;                 // 6250
    int gemm_blocks = (row_tiles + 7) / 8;         // 782
    k_gemm_wmma<<<gemm_blocks, T, 0, stream>>>(x, W, h, n);

    // 3) out = h * dinv^2  (self-loop contribution; fully initializes d_out)
    int total = n * HID;
    k_self<<<(total + T - 1) / T, T, 0, stream>>>(h, deg, out, total);

    // 4) scatter-add over edges, one wave (32 lanes x float4) per edge
    long long sthreads = (long long)E * 32;
    k_scatter<<<(int)((sthreads + T - 1) / T), T, 0, stream>>>(src, dst, h, deg, out, E);

    // 5) bias + PReLU
    k_final<<<(total + T - 1) / T, T, 0, stream>>>(out, bias, prelu_a, total);
}